// SplineconvModel_8169027797597
// MI455X (gfx1250) — compile-verified
//
#include <hip/hip_runtime.h>

// SplineConv (K=2, dim=1, degree=1) two-layer GNN for MI455X (gfx1250).
// Dense per-node transforms via V_WMMA_F32_16X16X4_F32 (fp32-accurate matrix path),
// edge lerp+scatter-mean via coalesced lane=channel gathers and f32 L2 atomics.

#define IN_F   128
#define H_F    32
#define OUT_F  64
#define NC1    96    // [W1[0] | W1[1] | root1] columns
#define NC2    192   // [W2[0] | W2[1] | root2] columns

typedef __attribute__((ext_vector_type(2))) float v2f;
typedef __attribute__((ext_vector_type(8))) float v8f;

// ---------------------------------------------------------------------------
// Zero-fill (capture-safe replacement for memset; agg/deg must start at 0)
// ---------------------------------------------------------------------------
__global__ void zero_f32(float* __restrict__ p, size_t n) {
    size_t t = (size_t)blockIdx.x * blockDim.x + threadIdx.x;
    if (t < n) p[t] = 0.0f;
}

// ---------------------------------------------------------------------------
// Pack [K,IN,H] weights + root into row-major B matrices:
//   B1[128][96]  = [W1[0] | W1[1] | root1]
//   B2[ 32][192] = [W2[0] | W2[1] | root2]
// ---------------------------------------------------------------------------
__global__ void pack_weights(const float* __restrict__ W1, const float* __restrict__ root1,
                             const float* __restrict__ W2, const float* __restrict__ root2,
                             float* __restrict__ B1, float* __restrict__ B2) {
    int t = blockIdx.x * blockDim.x + threadIdx.x;
    if (t < IN_F * NC1) {
        int i = t / NC1, j = t % NC1;
        float v;
        if (j < H_F)            v = W1[(0 * IN_F + i) * H_F + j];
        else if (j < 2 * H_F)   v = W1[(1 * IN_F + i) * H_F + (j - H_F)];
        else                    v = root1[i * H_F + (j - 2 * H_F)];
        B1[t] = v;
    }
    int t2 = t - IN_F * NC1;
    if (t2 >= 0 && t2 < H_F * NC2) {
        int i = t2 / NC2, j = t2 % NC2;
        float v;
        if (j < OUT_F)          v = W2[(0 * H_F + i) * OUT_F + j];
        else if (j < 2 * OUT_F) v = W2[(1 * H_F + i) * OUT_F + (j - OUT_F)];
        else                    v = root2[i * OUT_F + (j - 2 * OUT_F)];
        B2[t2] = v;
    }
}

// ---------------------------------------------------------------------------
// FP32 WMMA GEMM: C[M,NCOLS] = A[M,KDIM] * B[KDIM,NCOLS], row-major.
// One wave -> one 16x16 output tile; K-loop of V_WMMA_F32_16X16X4_F32.
// Grids divide exactly (M%16==0, NCOLS%16==0, tiles%waves_per_block==0)
// so there is no divergence: EXEC stays all-ones as WMMA requires.
//
// ISA layouts (CDNA5 §7.12.2):
//   A 16x4 : lane = M + 16*(K>>1), vgpr = K&1
//   B 4x16 : lane = N + 16*(K>>1), vgpr = K&1   (row-striped, half-wave K split)
//   C 16x16: vgpr r, lanes 0-15 -> M=r, lanes 16-31 -> M=r+8; N = lane&15
// ---------------------------------------------------------------------------
template <int KDIM, int NCOLS>
__global__ void wmma_gemm_f32(const float* __restrict__ A,
                              const float* __restrict__ B,
                              float* __restrict__ C) {
    const int lane = threadIdx.x & 31;
    const int wave = threadIdx.x >> 5;
    const int tile = blockIdx.x * (blockDim.x >> 5) + wave;

    constexpr int ntc = NCOLS / 16;
    const int m0 = (tile / ntc) * 16;
    const int n0 = (tile % ntc) * 16;

    const int l15   = lane & 15;
    const int khalf = (lane >> 4) << 1;      // 0 for lanes 0-15, 2 for lanes 16-31
    const long long arow = (long long)(m0 + l15) * KDIM;

    v8f acc = {};
#pragma unroll
    for (int kb = 0; kb < KDIM; kb += 4) {
        v2f a, b;
        a.x = A[arow + kb + khalf + 0];
        a.y = A[arow + kb + khalf + 1];
        b.x = B[(kb + khalf + 0) * NCOLS + n0 + l15];
        b.y = B[(kb + khalf + 1) * NCOLS + n0 + l15];
        acc = __builtin_amdgcn_wmma_f32_16x16x4_f32(
            /*neg_a=*/false, a, /*neg_b=*/false, b,
            /*c_mod=*/(short)0, acc, /*reuse_a=*/false, /*reuse_b=*/false);
    }

    const int rbase = m0 + ((lane >> 4) << 3);   // +8 rows for upper half-wave
    const long long cidx = (long long)rbase * NCOLS + n0 + l15;
#pragma unroll
    for (int r = 0; r < 8; ++r)
        C[cidx + (long long)r * NCOLS] = acc[r];
}

// ---------------------------------------------------------------------------
// Edge stage, layer 1: lane = channel (32 ch). K=2 => i0==0, frac==u.
// msg = xw[src,ch]*(1-u) + xw[src,32+ch]*u ; atomic scatter into agg1[dst].
// deg counted once per edge (reused by layer 2).
// ---------------------------------------------------------------------------
__global__ void edge_pass1(const int* __restrict__ src, const int* __restrict__ dst,
                           const float* __restrict__ u, const float* __restrict__ xw,
                           float* __restrict__ agg, float* __restrict__ deg, int E) {
    long long t = (long long)blockIdx.x * blockDim.x + threadIdx.x;
    int e = (int)(t >> 5);
    int c = (int)(t & 31);
    if (e >= E) return;
    int s = src[e], d = dst[e];
    float f = u[e];
    long long rb = (long long)s * NC1;
    float m0 = xw[rb + c];
    float m1 = xw[rb + H_F + c];
    float msg = m0 * (1.0f - f) + m1 * f;
    atomicAdd(&agg[(long long)d * H_F + c], msg);
    if (c == 0) atomicAdd(&deg[d], 1.0f);
}

// Node stage, layer 1: mean + root + bias + ReLU -> h[N,32]
__global__ void node_pass1(const float* __restrict__ agg, const float* __restrict__ deg,
                           const float* __restrict__ xw, const float* __restrict__ b1,
                           float* __restrict__ h, int N) {
    long long t = (long long)blockIdx.x * blockDim.x + threadIdx.x;
    int n = (int)(t >> 5);
    int c = (int)(t & 31);
    if (n >= N) return;
    float dg = deg[n];
    float inv = 1.0f / fmaxf(dg, 1.0f);
    float v = agg[(long long)n * H_F + c] * inv
            + xw[(long long)n * NC1 + 2 * H_F + c]   // x @ root1
            + b1[c];
    h[(long long)n * H_F + c] = fmaxf(v, 0.0f);
}

// Edge stage, layer 2: lane group of 64 = channel (64 ch).
__global__ void edge_pass2(const int* __restrict__ src, const int* __restrict__ dst,
                           const float* __restrict__ u, const float* __restrict__ hw,
                           float* __restrict__ agg, int E) {
    long long t = (long long)blockIdx.x * blockDim.x + threadIdx.x;
    int e = (int)(t >> 6);
    int c = (int)(t & 63);
    if (e >= E) return;
    int s = src[e], d = dst[e];
    float f = u[e];
    long long rb = (long long)s * NC2;
    float m0 = hw[rb + c];
    float m1 = hw[rb + OUT_F + c];
    float msg = m0 * (1.0f - f) + m1 * f;
    atomicAdd(&agg[(long long)d * OUT_F + c], msg);
}

// Node stage, layer 2: mean + root + bias -> out[N,64]
__global__ void node_pass2(const float* __restrict__ agg, const float* __restrict__ deg,
                           const float* __restrict__ hw, const float* __restrict__ b2,
                           float* __restrict__ out, int N) {
    long long t = (long long)blockIdx.x * blockDim.x + threadIdx.x;
    int n = (int)(t >> 6);
    int c = (int)(t & 63);
    if (n >= N) return;
    float dg = deg[n];
    float inv = 1.0f / fmaxf(dg, 1.0f);
    out[(long long)n * OUT_F + c] =
        agg[(long long)n * OUT_F + c] * inv
        + hw[(long long)n * NC2 + 2 * OUT_F + c]     // h @ root2
        + b2[c];
}

// ---------------------------------------------------------------------------
extern "C" void kernel_launch(void* const* d_in, const int* in_sizes, int n_in,
                              void* d_out, int out_size, void* d_ws, size_t ws_size,
                              hipStream_t stream) {
    const float* x     = (const float*)d_in[0];
    const int*   ei    = (const int*)  d_in[1];
    const float* eattr = (const float*)d_in[2];
    const float* W1    = (const float*)d_in[3];
    const float* root1 = (const float*)d_in[4];
    const float* b1    = (const float*)d_in[5];
    const float* W2    = (const float*)d_in[6];
    const float* root2 = (const float*)d_in[7];
    const float* b2    = (const float*)d_in[8];

    const int N = in_sizes[0] / IN_F;     // 200000 (multiple of 16)
    const int E = in_sizes[2];            // 3200000
    const int* srcI = ei;
    const int* dstI = ei + E;

    // Workspace carve-up (fp32), ~334 MB total.
    float* ws  = (float*)d_ws;
    float* B1  = ws;  ws += IN_F * NC1;
    float* B2  = ws;  ws += H_F * NC2;
    float* xw  = ws;  ws += (size_t)N * NC1;   // [x@W1[0] | x@W1[1] | x@root1]
    float* h   = ws;  ws += (size_t)N * H_F;
    float* hw  = ws;  ws += (size_t)N * NC2;   // [h@W2[0] | h@W2[1] | h@root2]
    float* agg1 = ws; ws += (size_t)N * H_F;
    float* agg2 = ws; ws += (size_t)N * OUT_F; // agg1|agg2|deg contiguous
    float* deg  = ws; ws += N;

    // 1) zero accumulators (deterministic every call)
    size_t nz = (size_t)N * (H_F + OUT_F + 1);
    zero_f32<<<(unsigned)((nz + 255) / 256), 256, 0, stream>>>(agg1, nz);

    // 2) pack weights
    {
        int tot = IN_F * NC1 + H_F * NC2;
        pack_weights<<<(tot + 255) / 256, 256, 0, stream>>>(W1, root1, W2, root2, B1, B2);
    }

    // 3) layer-1 dense: xw = x @ B1   ([N,128]x[128,96])
    {
        int tiles = (N / 16) * (NC1 / 16);            // 75000
        wmma_gemm_f32<IN_F, NC1><<<tiles / 8, 256, 0, stream>>>(x, B1, xw);
    }

    // 4) layer-1 edges + nodes
    {
        long long t1 = (long long)E * 32;
        edge_pass1<<<(unsigned)((t1 + 255) / 256), 256, 0, stream>>>(
            srcI, dstI, eattr, xw, agg1, deg, E);
        long long t2 = (long long)N * 32;
        node_pass1<<<(unsigned)((t2 + 255) / 256), 256, 0, stream>>>(
            agg1, deg, xw, b1, h, N);
    }

    // 5) layer-2 dense: hw = h @ B2   ([N,32]x[32,192])
    {
        int tiles = (N / 16) * (NC2 / 16);            // 150000
        wmma_gemm_f32<H_F, NC2><<<tiles / 8, 256, 0, stream>>>(h, B2, hw);
    }

    // 6) layer-2 edges + nodes -> d_out
    {
        long long t1 = (long long)E * 64;
        edge_pass2<<<(unsigned)((t1 + 255) / 256), 256, 0, stream>>>(
            srcI, dstI, eattr, hw, agg2, E);
        long long t2 = (long long)N * 64;
        node_pass2<<<(unsigned)((t2 + 255) / 256), 256, 0, stream>>>(
            agg2, deg, hw, b2, (float*)d_out, N);
    }
}